// GraphSAGE_44306882625537
// MI455X (gfx1250) — compile-verified
//
#include <hip/hip_runtime.h>

// ---------------------------------------------------------------------------
// GraphSAGE (2x SAGEConv, mean aggregation) for MI455X / gfx1250, wave32.
// Aggregation: fp32 global atomics (agg buffers fit the 192MB L2).
// GEMMs: V_WMMA_F32_16X16X4_F32 (full fp32 precision, matrix pipe),
//        shapes templated so the K-loop fully unrolls into load+wmma clauses.
// ---------------------------------------------------------------------------

typedef __attribute__((ext_vector_type(2))) float v2f;
typedef __attribute__((ext_vector_type(8))) float v8f;

__global__ void zero_f4_kernel(float4* __restrict__ p, int n4) {
    int i = blockIdx.x * blockDim.x + threadIdx.x;
    if (i < n4) p[i] = make_float4(0.f, 0.f, 0.f, 0.f);
}

__global__ void degree_kernel(const long long* __restrict__ dst,
                              float* __restrict__ deg, int E) {
    int e = blockIdx.x * blockDim.x + threadIdx.x;
    if (e < E) atomicAdd(deg + dst[e], 1.0f);
}

// One wave (32 lanes) per edge: lane covers D/32 contiguous floats of the row.
// Coalesced float4 row loads, coalesced f32 atomic adds into the dst row.
template <int D>
__global__ void scatter_add_kernel(const float* __restrict__ feat,
                                   const long long* __restrict__ src,
                                   const long long* __restrict__ dst,
                                   float* __restrict__ agg, int E) {
    constexpr int VPL = D / 32;  // floats per lane (4 or 8)
    int gid  = blockIdx.x * blockDim.x + threadIdx.x;
    int e    = gid >> 5;
    int lane = gid & 31;
    if (e >= E) return;  // uniform per wave
    const float* fr = feat + (size_t)src[e] * D + lane * VPL;
    float*       ar = agg  + (size_t)dst[e] * D + lane * VPL;
#pragma unroll
    for (int i = 0; i < VPL; i += 4) {
        float4 v = *(const float4*)(fr + i);
        atomicAdd(ar + i + 0, v.x);
        atomicAdd(ar + i + 1, v.y);
        atomicAdd(ar + i + 2, v.z);
        atomicAdd(ar + i + 3, v.w);
    }
}

// out[m,n] = act( (agg[m,:]/max(deg[m],1)) @ Wl + xr[m,:] @ Wr + bias[n] )
// One wave per 16x16 tile; block = 8 waves -> 16 rows x 128 cols.
// grid.x = Nnodes/16, grid.y = Dout/128. Requires Nnodes%16==0 (100000/16=6250).
// K, Dout compile-time: inner loop is immediate-offset loads + v_wmma only.
template <int K, int Dout, bool RELU>
__global__ void sage_gemm_wmma(const float* __restrict__ agg,
                               const float* __restrict__ deg,
                               const float* __restrict__ Wl,
                               const float* __restrict__ xr,
                               const float* __restrict__ Wr,
                               const float* __restrict__ bias,
                               float* __restrict__ out) {
    const int lane = threadIdx.x & 31;
    const int wave = threadIdx.x >> 5;       // 0..7
    const int half = lane >> 4;              // 0: K+{0,1}, 1: K+{2,3}
    const int l16  = lane & 15;
    const int mrow = blockIdx.x * 16 + l16;  // A row this lane feeds
    const int ncol = blockIdx.y * 128 + wave * 16 + l16;  // B/D column

    const float rdeg = 1.0f / fmaxf(deg[mrow], 1.0f);
    const v2f* arow_l = (const v2f*)(agg + (size_t)mrow * K) + half;
    const v2f* arow_r = (const v2f*)(xr  + (size_t)mrow * K) + half;
    const float* wl   = Wl + (size_t)(2 * half) * Dout + ncol;
    const float* wr   = Wr + (size_t)(2 * half) * Dout + ncol;

    v8f c = {};
    // pass 1: (agg * rdeg) @ Wl
#pragma unroll
    for (int k = 0; k < K; k += 4) {
        v2f a = arow_l[k >> 1];
        a.x *= rdeg;
        a.y *= rdeg;
        v2f b;
        b.x = wl[(size_t)k * Dout];
        b.y = wl[(size_t)(k + 1) * Dout];
        c = __builtin_amdgcn_wmma_f32_16x16x4_f32(false, a, false, b,
                                                  (short)0, c, false, false);
    }
    // pass 2: xr @ Wr (accumulate into same C)
#pragma unroll
    for (int k = 0; k < K; k += 4) {
        v2f a = arow_r[k >> 1];
        v2f b;
        b.x = wr[(size_t)k * Dout];
        b.y = wr[(size_t)(k + 1) * Dout];
        c = __builtin_amdgcn_wmma_f32_16x16x4_f32(false, a, false, b,
                                                  (short)0, c, false, false);
    }

    const float bv = bias[ncol];
    float* orow = out + (size_t)(blockIdx.x * 16 + 8 * half) * Dout + ncol;
#pragma unroll
    for (int r = 0; r < 8; ++r) {
        float v = c[r] + bv;
        if (RELU) v = fmaxf(v, 0.0f);
        orow[(size_t)r * Dout] = v;  // C/D layout: M = r + 8*half
    }
}

extern "C" void kernel_launch(void* const* d_in, const int* in_sizes, int n_in,
                              void* d_out, int out_size, void* d_ws, size_t ws_size,
                              hipStream_t stream) {
    const float*     x   = (const float*)d_in[0];
    const long long* ei  = (const long long*)d_in[1];  // int64 [2, E]
    const float*     Wl0 = (const float*)d_in[2];
    const float*     bl0 = (const float*)d_in[3];
    const float*     Wr0 = (const float*)d_in[4];
    const float*     Wl1 = (const float*)d_in[5];
    const float*     bl1 = (const float*)d_in[6];
    const float*     Wr1 = (const float*)d_in[7];

    const int DIN = 128, DHID = 256, DOUT = 128;
    const int N = in_sizes[0] / DIN;   // 100000
    const int E = in_sizes[1] / 2;     // 1600000
    const long long* srcI = ei;
    const long long* dstI = ei + E;

    // Workspace: deg[N] | buf1[N*256] (agg, both layers) | h[N*256]
    float* deg  = (float*)d_ws;
    float* buf1 = deg + (size_t)N;
    float* h    = buf1 + (size_t)N * DHID;
    float* outp = (float*)d_out;

    const int ethreads = 256;
    const int eblocks_scatter = (int)(((size_t)E * 32 + ethreads - 1) / ethreads);

    // --- Layer 0 ---
    {   // zero deg + agg0 (contiguous: N + N*128 floats)
        size_t n4 = (size_t)N * (1 + DIN) / 4;
        zero_f4_kernel<<<(int)((n4 + 255) / 256), 256, 0, stream>>>((float4*)deg, (int)n4);
    }
    degree_kernel<<<(E + 255) / 256, 256, 0, stream>>>(dstI, deg, E);
    scatter_add_kernel<128><<<eblocks_scatter, ethreads, 0, stream>>>(x, srcI, dstI, buf1, E);
    {
        dim3 grid(N / 16, DHID / 128);  // 6250 x 2
        sage_gemm_wmma<128, 256, true><<<grid, 256, 0, stream>>>(buf1, deg, Wl0, x,
                                                                 Wr0, bl0, h);
    }

    // --- Layer 1 ---
    {   // zero agg1 (N*256 floats)
        size_t n4 = (size_t)N * DHID / 4;
        zero_f4_kernel<<<(int)((n4 + 255) / 256), 256, 0, stream>>>((float4*)buf1, (int)n4);
    }
    scatter_add_kernel<256><<<eblocks_scatter, ethreads, 0, stream>>>(h, srcI, dstI, buf1, E);
    {
        dim3 grid(N / 16, DOUT / 128);  // 6250 x 1
        sage_gemm_wmma<256, 128, false><<<grid, 256, 0, stream>>>(buf1, deg, Wl1, h,
                                                                  Wr1, bl1, outp);
    }
}